// LossFunction_38938173505770
// MI455X (gfx1250) — compile-verified
//
#include <hip/hip_runtime.h>

typedef float v2f __attribute__((ext_vector_type(2)));
typedef float v8f __attribute__((ext_vector_type(8)));

constexpr int BDIM  = 256;           // 8 wave32 per block
constexpr int NBLK  = 1024;          // partials written to d_ws
constexpr int PAIRS = 128 * 32768;   // B * N = 4,194,304 box pairs

__device__ __forceinline__ float smooth_l1(float d) {
    // beta = 1: d < 1 ? 0.5*d*d : d - 0.5
    return (d < 1.0f) ? (0.5f * d * d) : (d - 0.5f);
}

__global__ __launch_bounds__(BDIM)
void iou_sl1_partial(const float4* __restrict__ t4,
                     const float4* __restrict__ p4,
                     float* __restrict__ partial) {
    const float inv_pairs = 1.0f / 4194304.0f;   // 1/(B*N)
    const float inv_elems = 1.0f / 16777216.0f;  // 1/(B*N*4)

    const int tid    = blockIdx.x * BDIM + threadIdx.x;
    const int stride = gridDim.x * BDIM;

    float acc = 0.0f;
#pragma unroll 4
    for (int i = tid; i < PAIRS; i += stride) {
        float4 t = t4[i];   // one box: global_load_b128, fully coalesced
        float4 p = p4[i];

        // matched-pair IoU
        float xx1 = fmaxf(t.x, p.x);
        float yy1 = fmaxf(t.y, p.y);
        float xx2 = fminf(t.z, p.z);
        float yy2 = fminf(t.w, p.w);
        float w   = fmaxf(xx2 - xx1, 0.0f);
        float h   = fmaxf(yy2 - yy1, 0.0f);
        float inter = w * h;
        float a1  = (t.z - t.x) * (t.w - t.y);
        float a2  = (p.z - p.x) * (p.w - p.y);
        float uni = a1 + a2 - inter;
        float iou = inter / uni;
        float l1  = fabsf(iou - 1.0f);

        // smooth_l1 over the 4 coords
        float sl = smooth_l1(fabsf(t.x - p.x))
                 + smooth_l1(fabsf(t.y - p.y))
                 + smooth_l1(fabsf(t.z - p.z))
                 + smooth_l1(fabsf(t.w - p.w));

        acc += l1 * inv_pairs + sl * inv_elems;   // pre-scaled single accumulator
    }

    // ---- wave32 reduction via V_WMMA_F32_16X16X4_F32 with all-ones B ----
    // A (16x4 f32, 2 VGPRs/lane): lane L<16 -> A[L][0]=acc, A[L][1]=0
    //                             lane L>=16 -> A[L-16][2]=acc, A[L-16][3]=0
    // B = ones(4x16)  =>  D[M][n] = acc[M] + acc[M+16]  (replicated over n)
    v2f a; a[0] = acc;  a[1] = 0.0f;
    v2f b; b[0] = 1.0f; b[1] = 1.0f;
    v8f c = {};
    c = __builtin_amdgcn_wmma_f32_16x16x4_f32(
            /*neg_a=*/false, a, /*neg_b=*/false, b,
            /*c_mod=*/(short)0, c, /*reuse_a=*/false, /*reuse_b=*/false);

    // each lane holds 8 D rows; lanes 0-15 sum rows 0-7, lanes 16-31 rows 8-15
    float s = c[0] + c[1] + c[2] + c[3] + c[4] + c[5] + c[6] + c[7];
    s += __shfl_xor(s, 16, 32);   // combine the two half-wave row groups

    __shared__ float lds[BDIM / 32];
    const int lane = threadIdx.x & 31;
    const int wid  = threadIdx.x >> 5;
    if (lane == 0) lds[wid] = s;
    __syncthreads();
    if (threadIdx.x == 0) {
        float bs = 0.0f;
#pragma unroll
        for (int i = 0; i < BDIM / 32; ++i) bs += lds[i];
        partial[blockIdx.x] = bs;
    }
}

__global__ __launch_bounds__(256)
void final_reduce(const float* __restrict__ partial, int n, float* __restrict__ out) {
    __shared__ float lds[256];
    float a = 0.0f;
    for (int i = threadIdx.x; i < n; i += 256) a += partial[i];
    lds[threadIdx.x] = a;
    __syncthreads();
#pragma unroll
    for (int off = 128; off > 0; off >>= 1) {
        if (threadIdx.x < off) lds[threadIdx.x] += lds[threadIdx.x + off];
        __syncthreads();
    }
    if (threadIdx.x == 0) out[0] = lds[0];
}

extern "C" void kernel_launch(void* const* d_in, const int* in_sizes, int n_in,
                              void* d_out, int out_size, void* d_ws, size_t ws_size,
                              hipStream_t stream) {
    const float4* t4 = (const float4*)d_in[0];  // targets [B,N,4] f32
    const float4* p4 = (const float4*)d_in[1];  // preds   [B,N,4] f32
    float* partial = (float*)d_ws;

    int nblk = NBLK;
    if (ws_size < (size_t)nblk * sizeof(float)) {
        nblk = (int)(ws_size / sizeof(float));
        if (nblk < 1) nblk = 1;
    }

    iou_sl1_partial<<<nblk, BDIM, 0, stream>>>(t4, p4, partial);
    final_reduce<<<1, 256, 0, stream>>>(partial, nblk, (float*)d_out);
}